// Hyper_Attention_Class_31172872635028
// MI455X (gfx1250) — compile-verified
//
#include <hip/hip_runtime.h>
#include <hip/hip_bf16.h>
#include <stdint.h>

typedef float v2f __attribute__((ext_vector_type(2)));
typedef float v4f __attribute__((ext_vector_type(4)));
typedef float v8f __attribute__((ext_vector_type(8)));

// ---------------- utility ----------------

__global__ void zero_i32(int* __restrict__ p, long n) {
  long i = (long)blockIdx.x * blockDim.x + threadIdx.x;
  long stride = (long)gridDim.x * blockDim.x;
  for (; i < n; i += stride) p[i] = 0;
}

__global__ void copy_i32(const int* __restrict__ s, int* __restrict__ d, int n) {
  int i = blockIdx.x * blockDim.x + threadIdx.x;
  if (i < n) d[i] = s[i];
}

// ---------------- layer-1 GEMM: XH[N,64] = X[N,K] @ W[K,64] (fp32 WMMA) ----------------
__global__ void gemm_x_w1(const float* __restrict__ X, const float* __restrict__ W,
                          float* __restrict__ XH, int n, int K) {
  const int lane = threadIdx.x & 31;
  const int wave = threadIdx.x >> 5;
  const int tile = blockIdx.x * (blockDim.x >> 5) + wave;
  const int m0 = tile * 16;
  if (m0 >= n) return;                       // wave-uniform: EXEC all-1 inside active waves

  const int mr = lane & 15;                  // A row / B col per ISA 16x4 fp32 layout
  const int kh = (lane >> 4) << 1;           // K pair select: 0 or 2

  v8f acc0 = {}, acc1 = {}, acc2 = {}, acc3 = {};
  const float* xr = X + (long)(m0 + mr) * K;

  for (int k0 = 0; k0 < K; k0 += 4) {
    v2f a = *(const v2f*)(xr + k0 + kh);     // contiguous pair -> b64 load
    const float* wp = W + (long)(k0 + kh) * 64 + mr;
    v2f b0; b0.x = wp[0];   b0.y = wp[64];
    v2f b1; b1.x = wp[16];  b1.y = wp[80];
    v2f b2; b2.x = wp[32];  b2.y = wp[96];
    v2f b3; b3.x = wp[48];  b3.y = wp[112];
    acc0 = __builtin_amdgcn_wmma_f32_16x16x4_f32(false, a, false, b0, (short)0, acc0, false, false);
    acc1 = __builtin_amdgcn_wmma_f32_16x16x4_f32(false, a, false, b1, (short)0, acc1, false, false);
    acc2 = __builtin_amdgcn_wmma_f32_16x16x4_f32(false, a, false, b2, (short)0, acc2, false, false);
    acc3 = __builtin_amdgcn_wmma_f32_16x16x4_f32(false, a, false, b3, (short)0, acc3, false, false);
  }

  const int nc = lane & 15;
  const int mb = m0 + ((lane >> 4) << 3);
  float* out = XH + (long)mb * 64 + nc;
#pragma unroll
  for (int r = 0; r < 8; ++r) {
    out[(long)r * 64 +  0] = acc0[r];
    out[(long)r * 64 + 16] = acc1[r];
    out[(long)r * 64 + 32] = acc2[r];
    out[(long)r * 64 + 48] = acc3[r];
  }
}

// ---------------- attention scalar scores s_i, s_j : [N, 8] ----------------
__global__ void si_sj(const float* __restrict__ xh, const float* __restrict__ att,
                      float* __restrict__ s_i, float* __restrict__ s_j, int n) {
  int i = blockIdx.x * blockDim.x + threadIdx.x;
  if (i >= n * 8) return;
  int node = i >> 3, h = i & 7;
  const float* xp = xh + (long)node * 64 + h * 8;
  const float* ai = att + h * 16;
  float si = 0.f, sj = 0.f;
#pragma unroll
  for (int c = 0; c < 8; ++c) { si += xp[c] * ai[c]; sj += xp[c] * ai[8 + c]; }
  s_i[i] = si; s_j[i] = sj;
}

// ---------------- CSR build ----------------
__global__ void seg_counts(const int* __restrict__ row, const int* __restrict__ col,
                           int* __restrict__ crow, int* __restrict__ ccol, int E) {
  int e = blockIdx.x * blockDim.x + threadIdx.x;
  if (e >= E) return;
  atomicAdd(&crow[row[e]], 1);
  atomicAdd(&ccol[col[e]], 1);
}

// single-workgroup exclusive scan over n counts; offs has n+1 entries
__global__ void exscan(const int* __restrict__ cnt, int* __restrict__ offs, int n) {
  __shared__ int sm[1024];
  int carry = 0;
  for (int base = 0; base < n; base += 1024) {
    int i = base + threadIdx.x;
    int v = (i < n) ? cnt[i] : 0;
    sm[threadIdx.x] = v;
    __syncthreads();
    for (int d = 1; d < 1024; d <<= 1) {
      int t = (threadIdx.x >= d) ? sm[threadIdx.x - d] : 0;
      __syncthreads();
      sm[threadIdx.x] += t;
      __syncthreads();
    }
    int incl  = sm[threadIdx.x];
    int total = sm[1023];
    if (i < n) offs[i] = carry + incl - v;
    carry += total;
    __syncthreads();
  }
  if (threadIdx.x == 0) offs[n] = carry;
}

// scatter edge ids + gather-endpoints into row/col buckets
__global__ void csr_fill(const int* __restrict__ row, const int* __restrict__ col,
                         int* __restrict__ cur_r, int* __restrict__ cur_c,
                         int* __restrict__ br_e, int* __restrict__ br_g,
                         int* __restrict__ bc_e, int* __restrict__ bc_g, int E) {
  int e = blockIdx.x * blockDim.x + threadIdx.x;
  if (e >= E) return;
  int r = row[e], c = col[e];
  int pr = atomicAdd(&cur_r[r], 1);
  br_e[pr] = e; br_g[pr] = c;
  int pc = atomicAdd(&cur_c[c], 1);
  bc_e[pc] = e; bc_g[pc] = r;
}

// ---------------- degree norms from CSR (no atomics) ----------------
__global__ void norms(const int* __restrict__ offs_r, const int* __restrict__ br_g,
                      const int* __restrict__ offs_c, const float* __restrict__ hw,
                      float* __restrict__ Dinv, float* __restrict__ Binv, int n) {
  int t = blockIdx.x * blockDim.x + threadIdx.x;
  if (t >= n) return;
  int b0 = offs_r[t], b1 = offs_r[t + 1];
  float d = 0.f;
  for (int k = b0; k < b1; ++k) d += hw[br_g[k]];
  Dinv[t] = d > 0.f ? 1.f / d : 0.f;
  int bc = offs_c[t + 1] - offs_c[t];
  Binv[t] = bc > 0 ? hw[t] / (float)bc : 0.f;
}

// ------- fused leaky-relu logit + segment softmax over row buckets -> alpha[E,8] -------
__global__ void segsoftmax_f(const int* __restrict__ offs_r, const int* __restrict__ br_e,
                             const int* __restrict__ br_g, const float* __restrict__ s_i,
                             const float* __restrict__ s_j, float* __restrict__ alpha, int n) {
  int idx = blockIdx.x * blockDim.x + threadIdx.x;
  int t = idx >> 3, h = idx & 7;
  if (t >= n) return;
  int b0 = offs_r[t], b1 = offs_r[t + 1];
  float si = s_i[t * 8 + h];
  float mx = -3.402823466e38f;
  for (int k = b0; k < b1; ++k) {
    float v = si + s_j[(long)br_g[k] * 8 + h];
    v = v >= 0.f ? v : 0.2f * v;
    mx = fmaxf(mx, v);
  }
  float s = 0.f;
  for (int k = b0; k < b1; ++k) {
    float v = si + s_j[(long)br_g[k] * 8 + h];
    v = v >= 0.f ? v : 0.2f * v;
    s += expf(v - mx);
  }
  float inv = 1.f / (s + 1e-16f);
  for (int k = b0; k < b1; ++k) {
    float v = si + s_j[(long)br_g[k] * 8 + h];
    v = v >= 0.f ? v : 0.2f * v;
    alpha[(long)br_e[k] * 8 + h] = expf(v - mx) * inv;
  }
}

// ---------------- propagate (64-ch, attention): float4 gather, zero atomics ----------------
// out[t, q*4..q*4+3] = norm[t] * sum_k xin[bg[k], q*4..] * alpha[be[k], q>>1]
__global__ void gather64(const int* __restrict__ offs, const int* __restrict__ bg,
                         const int* __restrict__ be, const float* __restrict__ xin,
                         const float* __restrict__ norm, const float* __restrict__ alpha,
                         float* __restrict__ outp, int n) {
  int idx = blockIdx.x * blockDim.x + threadIdx.x;
  int t = idx >> 4, q = idx & 15;          // 16 threads per node, 4 channels each
  if (t >= n) return;
  int b0 = offs[t], b1 = offs[t + 1];
  int h = q >> 1;                          // head = (q*4)/8
  v4f acc = {0.f, 0.f, 0.f, 0.f};
  for (int k = b0; k < b1; ++k) {
    if (((k - b0) & 31) == 0) {            // stream-ahead on bucket arrays
      __builtin_prefetch(bg + k + 128, 0, 1);
      __builtin_prefetch(be + k + 128, 0, 1);
    }
    int g = bg[k];
    float al = alpha[(long)be[k] * 8 + h];
    v4f xv = *(const v4f*)(xin + (long)g * 64 + q * 4);
    acc += xv * al;
  }
  float nm = norm[t];
  *(v4f*)(outp + (long)t * 64 + q * 4) = acc * nm;
}

__global__ void bias_elu(const float* __restrict__ in, const float* __restrict__ b,
                         float* __restrict__ out, long total) {
  long i = (long)blockIdx.x * blockDim.x + threadIdx.x;
  if (i >= total) return;
  float v = in[i] + b[i & 63];
  out[i] = v > 0.f ? v : (expf(v) - 1.f);
}

// ---------------- layer-2 GEMM: XH2[N,7] (stride 8) = H[N,64] @ W2[64,7] ----------------
__global__ void gemm_h_w2(const float* __restrict__ H, const float* __restrict__ W2,
                          float* __restrict__ XH2, int n) {
  int i = blockIdx.x * blockDim.x + threadIdx.x;
  if (i >= n) return;
  const float* hp = H + (long)i * 64;
  float acc[7] = {0.f, 0.f, 0.f, 0.f, 0.f, 0.f, 0.f};
  for (int k = 0; k < 64; ++k) {
    float hv = hp[k];
#pragma unroll
    for (int c = 0; c < 7; ++c) acc[c] += hv * W2[k * 7 + c];
  }
#pragma unroll
  for (int c = 0; c < 7; ++c) XH2[(long)i * 8 + c] = acc[c];
  XH2[(long)i * 8 + 7] = 0.f;
}

// ---------------- propagate (8-ch incl. zero pad, no attention): gather form ----------------
__global__ void gather8(const int* __restrict__ offs, const int* __restrict__ bg,
                        const float* __restrict__ xin, const float* __restrict__ norm,
                        float* __restrict__ outp, int n) {
  int idx = blockIdx.x * blockDim.x + threadIdx.x;
  int t = idx >> 3, j = idx & 7;
  if (t >= n) return;
  int b0 = offs[t], b1 = offs[t + 1];
  float acc = 0.f;
  for (int k = b0; k < b1; ++k) acc += xin[(long)bg[k] * 8 + j];
  outp[(long)t * 8 + j] = acc * norm[t];
}

__global__ void logsoftmax7(const float* __restrict__ in, const float* __restrict__ b2,
                            float* __restrict__ out, int n) {
  int i = blockIdx.x * blockDim.x + threadIdx.x;
  if (i >= n) return;
  float v[7];
  float mx = -3.402823466e38f;
#pragma unroll
  for (int c = 0; c < 7; ++c) { v[c] = in[(long)i * 8 + c] + b2[c]; mx = fmaxf(mx, v[c]); }
  float s = 0.f;
#pragma unroll
  for (int c = 0; c < 7; ++c) s += expf(v[c] - mx);
  float ls = logf(s);
#pragma unroll
  for (int c = 0; c < 7; ++c) out[(long)i * 7 + c] = v[c] - mx - ls;
}

// ---------------- driver ----------------

extern "C" void kernel_launch(void* const* d_in, const int* in_sizes, int n_in,
                              void* d_out, int out_size, void* d_ws, size_t ws_size,
                              hipStream_t stream) {
  const float* x    = (const float*)d_in[0];
  const int*   ei   = (const int*)d_in[1];
  const float* hw   = (const float*)d_in[2];
  const float* W1   = (const float*)d_in[3];
  const float* att1 = (const float*)d_in[4];
  const float* b1   = (const float*)d_in[5];
  const float* W2   = (const float*)d_in[6];
  const float* b2   = (const float*)d_in[7];
  float* out = (float*)d_out;

  const long N = in_sizes[2];            // 50000
  const long E = in_sizes[1] / 2;        // 1600000
  const int  K = (int)(in_sizes[0] / N); // 512
  const int* row = ei;
  const int* col = ei + E;

  // ---- workspace layout ----
  int* ip = (int*)d_ws;
  int* ccol   = ip;                 // N   (zeroed every call)
  int* crow   = ccol + N;           // N   (zeroed every call)
  int* offs_c = crow + N;           // N+1
  int* offs_r = offs_c + (N + 1);   // N+1
  int* cur_c  = offs_r + (N + 1);   // N
  int* cur_r  = cur_c + N;          // N
  int* bc_e   = cur_r + N;          // E  edge id,  bucketed by col
  int* bc_g   = bc_e + E;           // E  row endpoint, bucketed by col
  int* br_e   = bc_g + E;           // E  edge id,  bucketed by row
  int* br_g   = br_e + E;           // E  col endpoint, bucketed by row
  // 64-byte align the float region so float4 row accesses are 16B aligned
  float* fp   = (float*)(((uintptr_t)(br_g + E) + 63) & ~(uintptr_t)63);
  float* Dinv = fp;                 // N
  float* Binv = Dinv + N;           // N
  float* xh   = Binv + N;           // N*64  (later reused as out1)
  float* s_i  = xh + N * 64;        // N*8
  float* s_j  = s_i + N * 8;        // N*8
  float* out_e = s_j + N * 8;       // N*64  (later reused as hbuf)
  float* xh2  = out_e + N * 64;     // N*8
  float* out_e2 = xh2 + N * 8;      // N*8
  float* out2 = out_e2 + N * 8;     // N*8
  float* alpha = out2 + N * 8;      // E*8

  const int T = 256;
  zero_i32<<<512, T, 0, stream>>>(ccol, 2 * N);   // ccol + crow contiguous

  // dense part
  const long tiles = (N + 15) / 16;
  gemm_x_w1<<<(int)((tiles + 7) / 8), T, 0, stream>>>(x, W1, xh, (int)N, K);
  si_sj<<<(int)((N * 8 + T - 1) / T), T, 0, stream>>>(xh, att1, s_i, s_j, (int)N);

  // CSR build (only atomics in the whole pipeline)
  seg_counts<<<(int)((E + T - 1) / T), T, 0, stream>>>(row, col, crow, ccol, (int)E);
  exscan<<<1, 1024, 0, stream>>>(ccol, offs_c, (int)N);
  exscan<<<1, 1024, 0, stream>>>(crow, offs_r, (int)N);
  copy_i32<<<(int)((N + T - 1) / T), T, 0, stream>>>(offs_c, cur_c, (int)N);
  copy_i32<<<(int)((N + T - 1) / T), T, 0, stream>>>(offs_r, cur_r, (int)N);
  csr_fill<<<(int)((E + T - 1) / T), T, 0, stream>>>(row, col, cur_r, cur_c,
                                                     br_e, br_g, bc_e, bc_g, (int)E);

  // norms + attention (fused logit + segment softmax)
  norms<<<(int)((N + T - 1) / T), T, 0, stream>>>(offs_r, br_g, offs_c, hw, Dinv, Binv, (int)N);
  segsoftmax_f<<<(int)((N * 8 + T - 1) / T), T, 0, stream>>>(offs_r, br_e, br_g, s_i, s_j,
                                                             alpha, (int)N);

  // layer-1 propagate (float4 gather form)
  gather64<<<(int)((N * 16 + T - 1) / T), T, 0, stream>>>(offs_c, bc_g, bc_e, xh, Binv, alpha,
                                                          out_e, (int)N);
  gather64<<<(int)((N * 16 + T - 1) / T), T, 0, stream>>>(offs_r, br_g, br_e, out_e, Dinv, alpha,
                                                          xh /*out1 (xh dead)*/, (int)N);
  bias_elu<<<(int)((N * 64 + T - 1) / T), T, 0, stream>>>(xh, b1, out_e /*hbuf (dead)*/, N * 64);

  // layer 2 (reuses CSR + norms)
  gemm_h_w2<<<(int)((N + T - 1) / T), T, 0, stream>>>(out_e, W2, xh2, (int)N);
  gather8<<<(int)((N * 8 + T - 1) / T), T, 0, stream>>>(offs_c, bc_g, xh2, Binv, out_e2, (int)N);
  gather8<<<(int)((N * 8 + T - 1) / T), T, 0, stream>>>(offs_r, br_g, out_e2, Dinv, out2, (int)N);
  logsoftmax7<<<(int)((N + T - 1) / T), T, 0, stream>>>(out2, b2, out, (int)N);
}